// hhgnnConv_79242146611942
// MI455X (gfx1250) — compile-verified
//
#include <hip/hip_runtime.h>
#include <hip/hip_bf16.h>
#include <math.h>

#define N_V     100000
#define N_EDGE  20000
#define N_INC   500000
#define HH      8
#define CC      16
#define HC      128      // H*C
#define IN_DIM  64
#define SLOPE   0.2f

typedef __attribute__((ext_vector_type(2))) float v2f;
typedef __attribute__((ext_vector_type(8))) float v8f;

// ---- ordered-uint encoding for float atomic max (monotone for all finite floats) ----
__device__ __forceinline__ unsigned f2o(float f) {
    unsigned u = __float_as_uint(f);
    return (u & 0x80000000u) ? ~u : (u | 0x80000000u);
}
__device__ __forceinline__ float o2f(unsigned u) {
    return __uint_as_float((u & 0x80000000u) ? (u ^ 0x80000000u) : ~u);
}

__device__ __forceinline__ float lrelu(float x) { return x > 0.f ? x : SLOPE * x; }

// ------------------------------------------------------------------ fills
__global__ void fill_u32(unsigned* __restrict__ p, unsigned v, int n) {
    int i = blockIdx.x * blockDim.x + threadIdx.x;
    if (i < n) p[i] = v;
}

__global__ void scatter_cls(const int* __restrict__ idx, int n, int v, int* __restrict__ cls) {
    int i = blockIdx.x * blockDim.x + threadIdx.x;
    if (i < n) cls[idx[i]] = v;
}

// ------------------------------------------------------------------ GEMM via WMMA f32 16x16x4
// Xn[N_V,128] = X[N_V,64] @ W[64,128] + b[128]
// One wave per 16x16 output tile; W staged in LDS (32KB). N_V%16==0, tiles = 6250*8, 8 waves/block.
__global__ __launch_bounds__(256) void gemm_wmma(const float* __restrict__ X,
                                                 const float* __restrict__ W,
                                                 const float* __restrict__ bias,
                                                 float* __restrict__ Xn) {
    __shared__ float sW[IN_DIM * HC];            // 32 KB
    for (int i = threadIdx.x; i < IN_DIM * HC; i += 256) sW[i] = W[i];
    __syncthreads();

    const int wave = threadIdx.x >> 5;
    const int lane = threadIdx.x & 31;
    const int tile = blockIdx.x * 8 + wave;      // 50000 tiles total, grid = 6250 blocks
    const int rowTile = tile >> 3;               // 0..6249
    const int colTile = tile & 7;                // 0..7
    const int row0 = rowTile * 16;
    const int col0 = colTile * 16;
    const int half = lane >> 4;                  // 0: lanes 0-15, 1: lanes 16-31
    const int l    = lane & 15;

    const float* __restrict__ xrow = X + (size_t)(row0 + l) * IN_DIM;
    const float  bv_bias = bias[col0 + l];

    v8f acc = {};
    #pragma unroll
    for (int k0 = 0; k0 < IN_DIM; k0 += 4) {
        const int ka = k0 + half * 2;
        // A 16x4 f32: lane<16 -> {K=k0,k0+1}, lane>=16 -> {K=k0+2,k0+3}, M = l
        v2f a;
        a.x = xrow[ka];
        a.y = xrow[ka + 1];
        // B 4x16 f32: VGPR0 -> row K=ka, VGPR1 -> row K=ka+1, col = l
        v2f b;
        b.x = sW[ka * HC + col0 + l];
        b.y = sW[(ka + 1) * HC + col0 + l];
        acc = __builtin_amdgcn_wmma_f32_16x16x4_f32(false, a, false, b,
                                                    (short)0, acc, false, false);
    }
    // C/D: VGPR r -> row r (lanes 0-15) / row r+8 (lanes 16-31), col = l
    #pragma unroll
    for (int r = 0; r < 8; ++r) {
        int m = row0 + r + half * 8;
        Xn[(size_t)m * HC + col0 + l] = acc[r] + bv_bias;
    }
}

// ------------------------------------------------------------------ stage 1: vertex -> edge
__global__ void stage1_score(const float* __restrict__ Xn, const float* __restrict__ att_e,
                             const int* __restrict__ vertex, const int* __restrict__ edges,
                             float* __restrict__ beta, unsigned* __restrict__ emax) {
    int t = blockIdx.x * blockDim.x + threadIdx.x;
    if (t >= N_INC * HH) return;
    int e = t >> 3, h = t & 7;
    const float4* xr = (const float4*)(Xn + (size_t)vertex[e] * HC + h * CC);
    const float4* ar = (const float4*)(att_e + h * CC);
    float s = 0.f;
    #pragma unroll
    for (int i = 0; i < 4; ++i) {
        float4 x = xr[i], a = ar[i];
        s += x.x * a.x + x.y * a.y + x.z * a.z + x.w * a.w;
    }
    s = lrelu(s);
    beta[t] = s;
    atomicMax(&emax[edges[e] * HH + h], f2o(s));
}

__global__ void stage1_norm(float* __restrict__ beta, const unsigned* __restrict__ emax,
                            float* __restrict__ esum, const int* __restrict__ edges) {
    int t = blockIdx.x * blockDim.x + threadIdx.x;
    if (t >= N_INC * HH) return;
    int e = t >> 3, h = t & 7;
    float m  = o2f(emax[edges[e] * HH + h]);
    float ex = __expf(beta[t] - m);
    beta[t] = ex;
    atomicAdd(&esum[edges[e] * HH + h], ex);
}

__global__ void stage1_agg(const float* __restrict__ Xn, const float* __restrict__ beta,
                           const float* __restrict__ esum, const int* __restrict__ vertex,
                           const int* __restrict__ edges, float* __restrict__ Xe) {
    int t = blockIdx.x * blockDim.x + threadIdx.x;
    if (t >= N_INC * HH) return;
    int e = t >> 3, h = t & 7;
    float w = beta[t] / esum[edges[e] * HH + h];
    const float* xr = Xn + (size_t)vertex[e] * HC + h * CC;
    float*       xe = Xe + (size_t)edges[e] * HC + h * CC;
    #pragma unroll
    for (int c = 0; c < CC; ++c) atomicAdd(&xe[c], xr[c] * w);
}

// ------------------------------------------------------------------ stage 2: edge -> vertex
__global__ void stage2_score(const float* __restrict__ Xe,
                             const float* __restrict__ attA, const float* __restrict__ attU,
                             const float* __restrict__ attI, const int* __restrict__ cls,
                             const int* __restrict__ vertex, const int* __restrict__ edges,
                             float* __restrict__ alpha, unsigned* __restrict__ vmax) {
    int t = blockIdx.x * blockDim.x + threadIdx.x;
    if (t >= N_INC * HH) return;
    int e = t >> 3, h = t & 7;
    int c = cls[e];
    const float* av = (c == 0) ? attA : ((c == 1) ? attU : attI);
    const float4* xr = (const float4*)(Xe + (size_t)edges[e] * HC + h * CC);
    const float4* ar = (const float4*)(av + h * CC);
    float s = 0.f;
    #pragma unroll
    for (int i = 0; i < 4; ++i) {
        float4 x = xr[i], a = ar[i];
        s += x.x * a.x + x.y * a.y + x.z * a.z + x.w * a.w;
    }
    s = lrelu(s);
    alpha[t] = s;
    atomicMax(&vmax[vertex[e] * HH + h], f2o(s));
}

__global__ void stage2_norm(float* __restrict__ alpha, const unsigned* __restrict__ vmax,
                            float* __restrict__ vsum, const int* __restrict__ vertex) {
    int t = blockIdx.x * blockDim.x + threadIdx.x;
    if (t >= N_INC * HH) return;
    int e = t >> 3, h = t & 7;
    float m  = o2f(vmax[vertex[e] * HH + h]);
    float ex = __expf(alpha[t] - m);
    alpha[t] = ex;
    atomicAdd(&vsum[vertex[e] * HH + h], ex);
}

__global__ void stage2_agg(const float* __restrict__ Xe, const float* __restrict__ alpha,
                           const float* __restrict__ vsum, const int* __restrict__ vertex,
                           const int* __restrict__ edges, float* __restrict__ out) {
    int t = blockIdx.x * blockDim.x + threadIdx.x;
    if (t >= N_INC * HH) return;
    int e = t >> 3, h = t & 7;
    float w = alpha[t] / vsum[vertex[e] * HH + h];
    const float* xe = Xe + (size_t)edges[e] * HC + h * CC;
    float*       ov = out + (size_t)vertex[e] * HC + h * CC;
    #pragma unroll
    for (int c = 0; c < CC; ++c) atomicAdd(&ov[c], xe[c] * w);
}

__global__ void relu_out(float* __restrict__ out, int n) {
    int i = blockIdx.x * blockDim.x + threadIdx.x;
    if (i < n) out[i] = fmaxf(out[i], 0.f);
}

// ------------------------------------------------------------------ host side
static inline int cdiv(int a, int b) { return (a + b - 1) / b; }

extern "C" void kernel_launch(void* const* d_in, const int* in_sizes, int n_in,
                              void* d_out, int out_size, void* d_ws, size_t ws_size,
                              hipStream_t stream) {
    const float* X     = (const float*)d_in[0];
    const float* W_w   = (const float*)d_in[1];
    const float* W_b   = (const float*)d_in[2];
    const float* att_e = (const float*)d_in[3];
    const float* att_a = (const float*)d_in[4];
    const float* att_u = (const float*)d_in[5];
    const float* att_i = (const float*)d_in[6];
    const int*   vertex = (const int*)d_in[7];
    const int*   edges  = (const int*)d_in[8];
    // d_in[9] = V_class_index [E,H]: semantically equivalent to a per-incidence class label,
    // which we rebuild directly from idx_aspect/idx_user/idx_item.
    const int* idx_aspect = (const int*)d_in[10];
    const int* idx_user   = (const int*)d_in[11];
    const int* idx_item   = (const int*)d_in[12];
    const int n_aspect = in_sizes[10], n_user = in_sizes[11], n_item = in_sizes[12];

    // workspace layout (4-byte elements)
    float*    Xn    = (float*)d_ws;                              // N_V*128
    float*    Xe    = Xn   + (size_t)N_V * HC;                   // N_EDGE*128
    float*    beta  = Xe   + (size_t)N_EDGE * HC;                // E*H (reused as alpha)
    unsigned* emax  = (unsigned*)(beta + (size_t)N_INC * HH);    // N_EDGE*H
    float*    esum  = (float*)(emax + (size_t)N_EDGE * HH);      // N_EDGE*H
    unsigned* vmax  = (unsigned*)(esum + (size_t)N_EDGE * HH);   // N_V*H
    float*    vsum  = (float*)(vmax + (size_t)N_V * HH);         // N_V*H
    int*      cls   = (int*)(vsum + (size_t)N_V * HH);           // E
    float*    out   = (float*)d_out;

    const int B = 256;
    const int EH = N_INC * HH;

    // init (every call: graph-replay safe; 0u is both 0.0f and the ordered-min encoding)
    fill_u32<<<cdiv(N_EDGE * HC, B), B, 0, stream>>>((unsigned*)Xe, 0u, N_EDGE * HC);
    fill_u32<<<cdiv(N_EDGE * HH, B), B, 0, stream>>>(emax, 0u, N_EDGE * HH);
    fill_u32<<<cdiv(N_EDGE * HH, B), B, 0, stream>>>((unsigned*)esum, 0u, N_EDGE * HH);
    fill_u32<<<cdiv(N_V * HH, B), B, 0, stream>>>(vmax, 0u, N_V * HH);
    fill_u32<<<cdiv(N_V * HH, B), B, 0, stream>>>((unsigned*)vsum, 0u, N_V * HH);
    fill_u32<<<cdiv(out_size, B), B, 0, stream>>>((unsigned*)out, 0u, out_size);

    // class labels
    scatter_cls<<<cdiv(n_aspect, B), B, 0, stream>>>(idx_aspect, n_aspect, 0, cls);
    scatter_cls<<<cdiv(n_user,   B), B, 0, stream>>>(idx_user,   n_user,   1, cls);
    scatter_cls<<<cdiv(n_item,   B), B, 0, stream>>>(idx_item,   n_item,   2, cls);

    // dense projection via WMMA: 6250 blocks x 8 waves = 50000 16x16 tiles
    gemm_wmma<<<(N_V / 16) * 8 / 8, B, 0, stream>>>(X, W_w, W_b, Xn);

    // stage 1
    stage1_score<<<cdiv(EH, B), B, 0, stream>>>(Xn, att_e, vertex, edges, beta, emax);
    stage1_norm <<<cdiv(EH, B), B, 0, stream>>>(beta, emax, esum, edges);
    stage1_agg  <<<cdiv(EH, B), B, 0, stream>>>(Xn, beta, esum, vertex, edges, Xe);

    // stage 2 (alpha reuses beta buffer)
    stage2_score<<<cdiv(EH, B), B, 0, stream>>>(Xe, att_a, att_u, att_i, cls,
                                                vertex, edges, beta, vmax);
    stage2_norm <<<cdiv(EH, B), B, 0, stream>>>(beta, vmax, vsum, vertex);
    stage2_agg  <<<cdiv(EH, B), B, 0, stream>>>(Xe, beta, vsum, vertex, edges, out);

    relu_out<<<cdiv(out_size, B), B, 0, stream>>>(out, out_size);
}